// CancerGATE_36258113912901
// MI455X (gfx1250) — compile-verified
//
#include <hip/hip_runtime.h>
#include <hip/hip_bf16.h>
#include <math.h>

typedef float v2f __attribute__((ext_vector_type(2)));
typedef float v8f __attribute__((ext_vector_type(8)));
typedef int   v2i __attribute__((ext_vector_type(2)));

#define N_NODES 50000
#define N_EDGES 1600000
#define HEADS 4

#if __has_builtin(__builtin_amdgcn_global_load_async_to_lds_b64) && \
    __has_builtin(__builtin_amdgcn_s_wait_asynccnt)
#define HAVE_ASYNC_LDS 1
#else
#define HAVE_ASYNC_LDS 0
#endif

// ---------------------------------------------------------------------------
// Order-preserving float <-> uint encoding for atomicMax-based segment max
// ---------------------------------------------------------------------------
__device__ __forceinline__ unsigned fenc(float f) {
    unsigned u = __float_as_uint(f);
    return (u & 0x80000000u) ? ~u : (u | 0x80000000u);
}
__device__ __forceinline__ float fdec(unsigned k) {
    return (k & 0x80000000u) ? __uint_as_float(k & 0x7fffffffu)
                             : __uint_as_float(~k);
}

// ---------------------------------------------------------------------------
// WMMA f32 GEMM: C[M,N] = act(A[M,K] @ B + bias), optional B transposed.
// Block = 256 threads (8 wave32), tile 128(M) x 64(N), TK = 16.
// LDS layouts keep (k, k+1) pairs contiguous so each WMMA fragment is one
// aligned 8-byte LDS load straight into an even VGPR pair (no repack movs):
//   As[r][c]      : row stride 18 (even => 8B aligned; 9i mod 32 distinct)
//   Bs[kp][n][2]  : pair stride 160 words == 32 mod 64 => half-waves hit
//                   disjoint bank halves.
// A tile is staged with GLOBAL_LOAD_ASYNC_TO_LDS_B64 (ASYNCcnt) when the
// toolchain exposes it; next A chunk is prefetched (global_prefetch_b8).
// ACT: 0 = none, 1 = relu, 2 = sigmoid
// ---------------------------------------------------------------------------
template<int ACT, int TRANSB>
__global__ __launch_bounds__(256)
void gemm_wmma_f32(const float* __restrict__ A, int lda,
                   const float* __restrict__ B, int ldb,
                   const float* __restrict__ bias,
                   float* __restrict__ C, int ldc,
                   int M, int N, int K)
{
    __shared__ float As[128][18];
    __shared__ float Bs[8][80][2];   // [k>>1][n][k&1]

    const int tid  = threadIdx.x;
    const int wave = tid >> 5;
    const int lane = tid & 31;
    const int l15  = lane & 15;
    const int koff = (lane >> 4) * 2;   // hi half-wave holds K+2/K+3 per ISA layout

    const int m0 = blockIdx.y * 128;
    const int n0 = blockIdx.x * 64;

    v8f acc[4];
    #pragma unroll
    for (int j = 0; j < 4; ++j)
        #pragma unroll
        for (int v = 0; v < 8; ++v) acc[j][v] = 0.0f;

    for (int k0 = 0; k0 < K; k0 += 16) {
        // ---- stage A tile (128 x 16): async DMA to LDS, b64 per lane ----
#if HAVE_ASYNC_LDS
        #pragma unroll
        for (int i = 0; i < 4; ++i) {
            int idx = tid + i * 256;        // 0..1023
            int r   = idx >> 3;             // 0..127
            int c2  = (idx & 7) * 2;        // 0,2,..,14
            int gr  = m0 + r;
            if (gr < M) {
                __builtin_amdgcn_global_load_async_to_lds_b64(
                    (__attribute__((address_space(1))) v2i*)
                        (A + (size_t)gr * lda + (k0 + c2)),
                    (__attribute__((address_space(3))) v2i*)(&As[r][c2]),
                    0, 0);
            } else {
                As[r][c2]     = 0.0f;
                As[r][c2 + 1] = 0.0f;
            }
        }
#else
        #pragma unroll
        for (int i = 0; i < 8; ++i) {
            int idx = tid + i * 256;
            int r = idx >> 4, c = idx & 15;
            int gr = m0 + r;
            As[r][c] = (gr < M) ? A[(size_t)gr * lda + (k0 + c)] : 0.0f;
        }
#endif
        // ---- stage B tile (16 x 64) into pair-interleaved layout ----
        if (TRANSB == 0) {
            #pragma unroll
            for (int i = 0; i < 4; ++i) {
                int idx = tid + i * 256;
                int r = idx >> 6, c = idx & 63;
                Bs[r >> 1][c][r & 1] = B[(size_t)(k0 + r) * ldb + (n0 + c)];
            }
        } else {
            // B is [N,K] row-major; read column-contiguous for coalescing
            #pragma unroll
            for (int i = 0; i < 4; ++i) {
                int idx = tid + i * 256;
                int c = idx >> 4, r = idx & 15;
                Bs[r >> 1][c][r & 1] = B[(size_t)(n0 + c) * ldb + (k0 + r)];
            }
        }
#if HAVE_ASYNC_LDS
        __builtin_amdgcn_s_wait_asynccnt(0);
#endif
        __syncthreads();

        // ---- prefetch next A k-chunk while WMMAs run ----
        if (k0 + 16 < K) {
            int r  = tid >> 3;
            int gr = m0 + r;
            if (gr < M)
                __builtin_prefetch(A + (size_t)gr * lda + (k0 + 16) + ((tid & 7) * 2), 0, 0);
        }

        const int arow = wave * 16 + l15;
        #pragma unroll
        for (int kk = 0; kk < 16; kk += 4) {
            v2f a = *(const v2f*)(&As[arow][kk + koff]);          // ds_load_b64
            #pragma unroll
            for (int j = 0; j < 4; ++j) {
                v2f b = *(const v2f*)(&Bs[(kk + koff) >> 1][j * 16 + l15][0]);
                acc[j] = __builtin_amdgcn_wmma_f32_16x16x4_f32(
                    false, a, false, b, (short)0, acc[j], false, false);
            }
        }
        __syncthreads();
    }

    // ---- epilogue: C/D layout VGPR v -> M = v + 8*(lane>=16), N = lane&15 ----
    const int rbase = m0 + wave * 16 + (lane >> 4) * 8;
    #pragma unroll
    for (int j = 0; j < 4; ++j) {
        int col = n0 + j * 16 + l15;
        float bv = bias ? bias[col] : 0.0f;
        #pragma unroll
        for (int v = 0; v < 8; ++v) {
            int row = rbase + v;
            if (row < M) {
                float x = acc[j][v] + bv;
                if (ACT == 1)      x = x > 0.0f ? x : 0.0f;
                else if (ACT == 2) x = 1.0f / (1.0f + expf(-x));
                C[(size_t)row * ldc + col] = x;
            }
        }
    }
}

// ---------------------------------------------------------------------------
// el[n,h] = sum_d z[n,h,d]*al[h,d] ; er likewise. One wave per (node,head).
// ---------------------------------------------------------------------------
__global__ __launch_bounds__(256)
void attn_score_kernel(const float* __restrict__ z,
                       const float* __restrict__ al,
                       const float* __restrict__ ar,
                       float* __restrict__ el, float* __restrict__ er,
                       int Nn, int D)
{
    int gtid = blockIdx.x * blockDim.x + threadIdx.x;
    int wid  = gtid >> 5;
    int lane = gtid & 31;
    if (wid >= Nn * HEADS) return;
    int n = wid >> 2, h = wid & 3;
    const float* zp  = z  + (size_t)n * (HEADS * D) + h * D;
    const float* alp = al + h * D;
    const float* arp = ar + h * D;
    float sl = 0.0f, sr = 0.0f;
    for (int d = lane; d < D; d += 32) {
        float zv = zp[d];
        sl += zv * alp[d];
        sr += zv * arp[d];
    }
    #pragma unroll
    for (int off = 16; off; off >>= 1) {
        sl += __shfl_xor(sl, off, 32);
        sr += __shfl_xor(sr, off, 32);
    }
    if (lane == 0) { el[wid] = sl; er[wid] = sr; }
}

// ---------------------------------------------------------------------------
// e = leaky_relu(el[src]+er[dst]); segment max into menc[dst] (u32 atomicMax)
// ---------------------------------------------------------------------------
__global__ __launch_bounds__(256)
void edge_max_kernel(const int* __restrict__ src, const int* __restrict__ dst,
                     const float* __restrict__ el, const float* __restrict__ er,
                     float* __restrict__ eval, unsigned* __restrict__ menc, int E)
{
    int idx = blockIdx.x * blockDim.x + threadIdx.x;
    if (idx >= E * HEADS) return;
    int e = idx >> 2, h = idx & 3;
    float x = el[src[e] * HEADS + h] + er[dst[e] * HEADS + h];
    x = (x > 0.0f) ? x : 0.2f * x;
    eval[idx] = x;
    atomicMax(&menc[dst[e] * HEADS + h], fenc(x));
}

// ex = exp(e - m[dst]); segment sum into s[dst]
__global__ __launch_bounds__(256)
void edge_exp_kernel(const int* __restrict__ dst, float* __restrict__ eval,
                     const unsigned* __restrict__ menc, float* __restrict__ s, int E)
{
    int idx = blockIdx.x * blockDim.x + threadIdx.x;
    if (idx >= E * HEADS) return;
    int e = idx >> 2, h = idx & 3;
    int d = dst[e] * HEADS + h;
    float ex = expf(eval[idx] - fdec(menc[d]));
    eval[idx] = ex;
    atomicAdd(&s[d], ex);
}

// rst[dst,h,:] += alpha * z[src,h,:]. One wave per (edge, head).
__global__ __launch_bounds__(256)
void edge_agg_kernel(const int* __restrict__ src, const int* __restrict__ dst,
                     const float* __restrict__ eval, const float* __restrict__ s,
                     const float* __restrict__ z, float* __restrict__ rst,
                     int E, int D)
{
    int gtid = blockIdx.x * blockDim.x + threadIdx.x;
    int wid  = gtid >> 5;
    int lane = gtid & 31;
    if (wid >= E * HEADS) return;
    int e = wid >> 2, h = wid & 3;
    int sn = src[e], dn = dst[e];
    float alpha = eval[wid] / (s[dn * HEADS + h] + 1e-9f);
    const float* zp = z   + (size_t)sn * (HEADS * D) + h * D;
    float*       rp = rst + (size_t)dn * (HEADS * D) + h * D;
    for (int d = lane; d < D; d += 32)
        atomicAdd(&rp[d], alpha * zp[d]);
}

// out[i] = relu(rst[i] + r[i])   (in-place into rst)
__global__ __launch_bounds__(256)
void add_relu_kernel(float* __restrict__ rst, const float* __restrict__ r, size_t n)
{
    size_t i = blockIdx.x * (size_t)blockDim.x + threadIdx.x;
    if (i < n) {
        float x = rst[i] + r[i];
        rst[i] = x > 0.0f ? x : 0.0f;
    }
}

// dm[n,j] = mean over 4 heads of d2[n, h*96 + j]
__global__ __launch_bounds__(256)
void mean_heads_kernel(const float* __restrict__ d2, float* __restrict__ dm, int Nn)
{
    int i = blockIdx.x * blockDim.x + threadIdx.x;
    if (i >= Nn * 96) return;
    int n = i / 96, j = i - n * 96;
    const float* p = d2 + (size_t)n * 384 + j;
    dm[i] = 0.25f * (p[0] + p[96] + p[192] + p[288]);
}

// ---------------------------------------------------------------------------
extern "C" void kernel_launch(void* const* d_in, const int* in_sizes, int n_in,
                              void* d_out, int out_size, void* d_ws, size_t ws_size,
                              hipStream_t stream) {
    const float* feat0 = (const float*)d_in[0];
    const float* feat1 = (const float*)d_in[1];
    const float* feat2 = (const float*)d_in[2];
    const int*   src   = (const int*)d_in[3];
    const int*   dst   = (const int*)d_in[4];
    const float* W_emb0 = (const float*)d_in[5];
    const float* b_emb0 = (const float*)d_in[6];
    const float* W_emb1 = (const float*)d_in[7];
    const float* b_emb1 = (const float*)d_in[8];
    const float* W_emb2 = (const float*)d_in[9];
    const float* b_emb2 = (const float*)d_in[10];
    const float* fc0  = (const float*)d_in[11];
    const float* al0  = (const float*)d_in[12];
    const float* ar0  = (const float*)d_in[13];
    const float* res0 = (const float*)d_in[14];
    const float* fc1  = (const float*)d_in[15];
    const float* al1  = (const float*)d_in[16];
    const float* ar1  = (const float*)d_in[17];
    const float* res1 = (const float*)d_in[18];
    const float* Wd0 = (const float*)d_in[19];
    const float* bd0 = (const float*)d_in[20];
    const float* Wd1 = (const float*)d_in[21];
    const float* bd1 = (const float*)d_in[22];
    const float* Wd2 = (const float*)d_in[23];
    const float* bd2 = (const float*)d_in[24];
    float* out = (float*)d_out;

    const size_t N = N_NODES;
    const int    E = N_EDGES;

    float* ws = (float*)d_ws;
    size_t off = 0;
    float* h0   = ws + off; off += N * 384;   // embed concat; reused as d2
    float* z0   = ws + off; off += N * 512;   // layer0 projection; reused as d1
    float* r0   = ws + off; off += N * 512;   // layer0 residual
    float* rst0 = ws + off; off += N * 512;   // layer0 agg; becomes h1 in-place
    float* z1   = ws + off; off += N * 256;
    float* r1   = ws + off; off += N * 256;
    float* rst1 = ws + off; off += N * 256;   // becomes relu(h2) in-place
    float* el   = ws + off; off += N * HEADS;
    float* er   = ws + off; off += N * HEADS;
    unsigned* menc = (unsigned*)(ws + off); off += N * HEADS;
    float* sden = ws + off; off += N * HEADS;
    float* eval = ws + off; off += (size_t)E * HEADS;
    float* dm   = ws + off; off += N * 96;

    dim3 blk(256);
    const int mblk = (N_NODES + 127) / 128;
    const int eh_blocks   = (E * HEADS + 255) / 256;            // thread per (e,h)
    const int nh_wblocks  = ((int)N * HEADS * 32 + 255) / 256;  // wave per (n,h)
    const int eh_wblocks  = ((long long)E * HEADS * 32 + 255) / 256; // wave per (e,h)

    // ---- embedding: 3x Dense(512->128) relu, concatenated into h0[N,384] ----
    {
        dim3 g(128 / 64, mblk);
        gemm_wmma_f32<1,0><<<g, blk, 0, stream>>>(feat0, 512, W_emb0, 128, b_emb0, h0 +   0, 384, N_NODES, 128, 512);
        gemm_wmma_f32<1,0><<<g, blk, 0, stream>>>(feat1, 512, W_emb1, 128, b_emb1, h0 + 128, 384, N_NODES, 128, 512);
        gemm_wmma_f32<1,0><<<g, blk, 0, stream>>>(feat2, 512, W_emb2, 128, b_emb2, h0 + 256, 384, N_NODES, 128, 512);
    }

    // ---- GAT layer 0: z0 = h0@fc0, r0 = h0@res0 (384 -> 512) ----
    {
        dim3 g(512 / 64, mblk);
        gemm_wmma_f32<0,0><<<g, blk, 0, stream>>>(h0, 384, fc0,  512, nullptr, z0, 512, N_NODES, 512, 384);
        gemm_wmma_f32<0,0><<<g, blk, 0, stream>>>(h0, 384, res0, 512, nullptr, r0, 512, N_NODES, 512, 384);
    }
    attn_score_kernel<<<nh_wblocks, blk, 0, stream>>>(z0, al0, ar0, el, er, N_NODES, 128);

    (void)hipMemsetAsync(menc, 0, N * HEADS * sizeof(unsigned), stream);
    (void)hipMemsetAsync(sden, 0, N * HEADS * sizeof(float), stream);
    (void)hipMemsetAsync(rst0, 0, N * 512 * sizeof(float), stream);
    edge_max_kernel<<<eh_blocks, blk, 0, stream>>>(src, dst, el, er, eval, menc, E);
    edge_exp_kernel<<<eh_blocks, blk, 0, stream>>>(dst, eval, menc, sden, E);
    edge_agg_kernel<<<eh_wblocks, blk, 0, stream>>>(src, dst, eval, sden, z0, rst0, E, 128);
    add_relu_kernel<<<(int)((N * 512 + 255) / 256), blk, 0, stream>>>(rst0, r0, N * 512); // h1 in rst0

    // ---- GAT layer 1: z1 = h1@fc1, r1 = h1@res1 (512 -> 256) ----
    {
        dim3 g(256 / 64, mblk);
        gemm_wmma_f32<0,0><<<g, blk, 0, stream>>>(rst0, 512, fc1,  256, nullptr, z1, 256, N_NODES, 256, 512);
        gemm_wmma_f32<0,0><<<g, blk, 0, stream>>>(rst0, 512, res1, 256, nullptr, r1, 256, N_NODES, 256, 512);
    }
    attn_score_kernel<<<nh_wblocks, blk, 0, stream>>>(z1, al1, ar1, el, er, N_NODES, 64);

    (void)hipMemsetAsync(menc, 0, N * HEADS * sizeof(unsigned), stream);
    (void)hipMemsetAsync(sden, 0, N * HEADS * sizeof(float), stream);
    (void)hipMemsetAsync(rst1, 0, N * 256 * sizeof(float), stream);
    edge_max_kernel<<<eh_blocks, blk, 0, stream>>>(src, dst, el, er, eval, menc, E);
    edge_exp_kernel<<<eh_blocks, blk, 0, stream>>>(dst, eval, menc, sden, E);
    edge_agg_kernel<<<eh_wblocks, blk, 0, stream>>>(src, dst, eval, sden, z1, rst1, E, 64);
    // decoder only ever consumes relu(h2), so fuse the relu here
    add_relu_kernel<<<(int)((N * 256 + 255) / 256), blk, 0, stream>>>(rst1, r1, N * 256);

    // ---- decoder: tied weights, transposed ----
    {   // d1 = relu(relu(h2) @ fc1^T)  [N,256]x[256,512]
        dim3 g(512 / 64, mblk);
        gemm_wmma_f32<1,1><<<g, blk, 0, stream>>>(rst1, 256, fc1, 256, nullptr, z0, 512, N_NODES, 512, 256);
    }
    {   // d2 = relu(d1 @ fc0^T)  [N,512]x[512,384]
        dim3 g(384 / 64, mblk);
        gemm_wmma_f32<1,1><<<g, blk, 0, stream>>>(z0, 512, fc0, 512, nullptr, h0, 384, N_NODES, 384, 512);
    }
    mean_heads_kernel<<<(int)((N * 96 + 255) / 256), blk, 0, stream>>>(h0, dm, N_NODES);

    // ---- 3 sigmoid output heads (96 -> 512), straight into d_out ----
    {
        dim3 g(512 / 64, mblk);
        gemm_wmma_f32<2,0><<<g, blk, 0, stream>>>(dm, 96, Wd0, 512, bd0, out,               512, N_NODES, 512, 96);
        gemm_wmma_f32<2,0><<<g, blk, 0, stream>>>(dm, 96, Wd1, 512, bd1, out + N * 512,     512, N_NODES, 512, 96);
        gemm_wmma_f32<2,0><<<g, blk, 0, stream>>>(dm, 96, Wd2, 512, bd2, out + 2 * N * 512, 512, N_NODES, 512, 96);
    }
}